// MyFCN_80822694576321
// MI455X (gfx1250) — compile-verified
//
#include <hip/hip_runtime.h>

// sum(left @ weight) + bias, with left:(8192,4096) f32, weight:(4096,4096) f32.
// Factorization: sum = dot(colsum(left), rowsum(weight)) + bias.
// Memory-bound: ~192 MB streamed once -> ~8.3 us at 23.3 TB/s.

typedef __attribute__((ext_vector_type(2))) float v2f;
typedef __attribute__((ext_vector_type(8))) float v8f;

#define M_ROWS 8192
#define KDIM   4096   // cols of left == rows of weight
#define NDIM   4096   // cols of weight
#define S_STRIPES 64
#define ROWS_PER_STRIPE (M_ROWS / S_STRIPES)   // 128

// ---------------------------------------------------------------------------
// K1: partial column sums of left. grid = (4 col-blocks, 64 row-stripes),
// 256 threads; each thread owns 4 consecutive columns via float4 loads.
// Per-column sums are layout-sensitive -> plain packed VALU adds.
// ---------------------------------------------------------------------------
__global__ __launch_bounds__(256) void colsum_partial_kernel(
    const float* __restrict__ left, float* __restrict__ partial) {
  const int c0 = blockIdx.x * 1024 + threadIdx.x * 4;   // column of first lane elem
  const int r0 = blockIdx.y * ROWS_PER_STRIPE;
  const float4* src =
      reinterpret_cast<const float4*>(left + (size_t)r0 * KDIM + c0);
  const int KV = KDIM / 4;  // row stride in float4 units

  float4 a0 = make_float4(0.f, 0.f, 0.f, 0.f);
  float4 a1 = make_float4(0.f, 0.f, 0.f, 0.f);
#pragma unroll 4
  for (int r = 0; r < ROWS_PER_STRIPE; r += 2) {
    float4 x = src[(size_t)r * KV];
    float4 y = src[(size_t)(r + 1) * KV];
    a0.x += x.x; a0.y += x.y; a0.z += x.z; a0.w += x.w;
    a1.x += y.x; a1.y += y.y; a1.z += y.z; a1.w += y.w;
  }
  float4 out = make_float4(a0.x + a1.x, a0.y + a1.y, a0.z + a1.z, a0.w + a1.w);
  reinterpret_cast<float4*>(partial + (size_t)blockIdx.y * KDIM + c0)[0] = out;
}

// ---------------------------------------------------------------------------
// K2: row sums of weight via WMMA. One wave per row (512 blocks x 8 waves).
// A = all-ones 16x4 f32 matrix; D += A x B means every D element accumulates a
// column-sum of B, so sum(all D) = 16 * sum(all B elements) -- independent of
// the exact B VGPR layout. Each iteration: one global_load_b128 + two
// v_wmma_f32_16x16x4_f32 folding 128 row elements.
// ---------------------------------------------------------------------------
__global__ __launch_bounds__(256) void rowsum_wmma_kernel(
    const float* __restrict__ w, float* __restrict__ rowsum) {
  const int lane = threadIdx.x & 31;
  const int wave = threadIdx.x >> 5;
  const int row  = blockIdx.x * 8 + wave;
  const float4* src = reinterpret_cast<const float4*>(w + (size_t)row * NDIM) + lane;

  const v2f ones = {1.0f, 1.0f};
  v8f acc = {};
#pragma unroll 4
  for (int i = 0; i < NDIM / 128; ++i) {      // 32 iterations
    float4 x = src[(size_t)i * 32];           // 128 floats per wave per iter
    v2f b0 = {x.x, x.y};
    v2f b1 = {x.z, x.w};
    acc = __builtin_amdgcn_wmma_f32_16x16x4_f32(false, ones, false, b0,
                                                (short)0, acc, false, false);
    acc = __builtin_amdgcn_wmma_f32_16x16x4_f32(false, ones, false, b1,
                                                (short)0, acc, false, false);
  }
  float t = acc[0] + acc[1] + acc[2] + acc[3] + acc[4] + acc[5] + acc[6] + acc[7];
#pragma unroll
  for (int off = 16; off > 0; off >>= 1) t += __shfl_xor(t, off, 32);
  if (lane == 0) rowsum[row] = t * (1.0f / 16.0f);
}

// ---------------------------------------------------------------------------
// K3: fold the 64 left-stripes into the final 4096 column sums.
// ---------------------------------------------------------------------------
__global__ __launch_bounds__(256) void stripe_reduce_kernel(
    const float* __restrict__ partial, float* __restrict__ colsum) {
  const int c0 = (blockIdx.x * 256 + threadIdx.x) * 4;
  float4 a = make_float4(0.f, 0.f, 0.f, 0.f);
#pragma unroll 4
  for (int s = 0; s < S_STRIPES; ++s) {
    float4 x = reinterpret_cast<const float4*>(partial + (size_t)s * KDIM + c0)[0];
    a.x += x.x; a.y += x.y; a.z += x.z; a.w += x.w;
  }
  reinterpret_cast<float4*>(colsum + c0)[0] = a;
}

// ---------------------------------------------------------------------------
// K4: dot(colsum, rowsum) + bias via the same WMMA-ones total reduction.
// One block (8 waves); each wave folds 512 elements in 4 load+2xWMMA steps.
// ---------------------------------------------------------------------------
__global__ __launch_bounds__(256) void dot_finish_kernel(
    const float* __restrict__ colsum, const float* __restrict__ rowsum,
    const float* __restrict__ bias, float* __restrict__ out) {
  __shared__ float wsum[8];
  const int lane = threadIdx.x & 31;
  const int wave = threadIdx.x >> 5;

  const v2f ones = {1.0f, 1.0f};
  v8f acc = {};
#pragma unroll
  for (int i = 0; i < 4; ++i) {
    const int base4 = (wave * 512 + i * 128) / 4 + lane;   // float4 index
    float4 c = reinterpret_cast<const float4*>(colsum)[base4];
    float4 r = reinterpret_cast<const float4*>(rowsum)[base4];
    v2f b0 = {c.x * r.x, c.y * r.y};
    v2f b1 = {c.z * r.z, c.w * r.w};
    acc = __builtin_amdgcn_wmma_f32_16x16x4_f32(false, ones, false, b0,
                                                (short)0, acc, false, false);
    acc = __builtin_amdgcn_wmma_f32_16x16x4_f32(false, ones, false, b1,
                                                (short)0, acc, false, false);
  }
  float t = acc[0] + acc[1] + acc[2] + acc[3] + acc[4] + acc[5] + acc[6] + acc[7];
#pragma unroll
  for (int off = 16; off > 0; off >>= 1) t += __shfl_xor(t, off, 32);
  if (lane == 0) wsum[wave] = t * (1.0f / 16.0f);
  __syncthreads();
  if (threadIdx.x == 0) {
    float s = wsum[0] + wsum[1] + wsum[2] + wsum[3] +
              wsum[4] + wsum[5] + wsum[6] + wsum[7];
    out[0] = s + bias[0];
  }
}

// ---------------------------------------------------------------------------
extern "C" void kernel_launch(void* const* d_in, const int* in_sizes, int n_in,
                              void* d_out, int out_size, void* d_ws, size_t ws_size,
                              hipStream_t stream) {
  const float* left   = (const float*)d_in[0];
  const float* weight = (const float*)d_in[1];
  const float* bias   = (const float*)d_in[2];
  float* out = (float*)d_out;

  // Workspace layout (floats): [0, 64*4096) stripe partials,
  // then 4096 row sums, then 4096 column sums.  ~1.03 MB total.
  float* ws      = (float*)d_ws;
  float* partial = ws;                              // 64*4096
  float* rowsum  = ws + (size_t)S_STRIPES * KDIM;   // 4096
  float* colsum  = rowsum + KDIM;                   // 4096

  colsum_partial_kernel<<<dim3(4, S_STRIPES), 256, 0, stream>>>(left, partial);
  rowsum_wmma_kernel<<<dim3(KDIM / 8), 256, 0, stream>>>(weight, rowsum);
  stripe_reduce_kernel<<<dim3(4), 256, 0, stream>>>(partial, colsum);
  dot_finish_kernel<<<dim3(1), 256, 0, stream>>>(colsum, rowsum, bias, out);
}